// AggrSum_48464410968234
// MI455X (gfx1250) — compile-verified
//
#include <hip/hip_runtime.h>

// ---------------------------------------------------------------------------
// segment_sum(H[N,128] f32, idx[N] i32) -> out[V,128] f32   (V*128 = out_size)
//
// Memory-bound scatter-add: ~375MB HBM traffic -> ~16us floor at 23.3 TB/s.
// One wave32 per row: 32 lanes x 16B = one 512B row, perfectly coalesced.
// CDNA5 path: global_load_async_to_lds_b128 (ASYNCcnt) with th:TH_LOAD_NT so
// the 320MB one-shot stream of H does not evict the 25.6MB output region from
// the 192MB L2, where all 80M global_atomic_add_f32 RMWs resolve.
// ---------------------------------------------------------------------------

typedef __attribute__((ext_vector_type(4))) float floatx4;

#define WAVE            32
#define WAVES_PER_BLOCK 8
#define BLOCK           (WAVE * WAVES_PER_BLOCK)
#define DCOLS           128   // feature dim, 512 bytes per row

__global__ __launch_bounds__(BLOCK) void AggrSum_scatter_kernel(
    const float* __restrict__ H,
    const int*   __restrict__ idx,
    float*       __restrict__ out,
    int nrows)
{
    // Double buffer: 2 x 8 waves x 512B = 8 KB LDS per block.
    __shared__ float buf[2][WAVES_PER_BLOCK][DCOLS];

    const int lane = threadIdx.x & (WAVE - 1);
    const int wv   = threadIdx.x >> 5;
    const int stride = gridDim.x * WAVES_PER_BLOCK;     // total waves in grid
    int row = blockIdx.x * WAVES_PER_BLOCK + wv;        // one row per wave

    // 32-bit LDS byte offsets of this lane's 16B slot in each buffer.
    // (Flat shared pointers carry the LDS offset in their low 32 bits.)
    const unsigned lds0 = (unsigned)(size_t)(const void*)&buf[0][wv][lane * 4];
    const unsigned lds1 = (unsigned)(size_t)(const void*)&buf[1][wv][lane * 4];

    // Prologue: kick off the async copy of this wave's first row.
    if (row < nrows) {
        unsigned goff = (unsigned)row * (DCOLS * 4u) + (unsigned)lane * 16u;
        asm volatile("global_load_async_to_lds_b128 %0, %1, %2 th:TH_LOAD_NT"
                     :: "v"(lds0), "v"(goff), "s"(H)
                     : "memory");
    }

    int b = 0;
    while (row < nrows) {                                // wave-uniform loop
        const int nxt = row + stride;
        if (nxt < nrows) {
            // Prefetch next row into the other buffer, then wait for current.
            unsigned goff = (unsigned)nxt * (DCOLS * 4u) + (unsigned)lane * 16u;
            asm volatile("global_load_async_to_lds_b128 %0, %1, %2 th:TH_LOAD_NT"
                         :: "v"(b ? lds0 : lds1), "v"(goff), "s"(H)
                         : "memory");
            asm volatile("s_wait_asynccnt 1" ::: "memory");  // current row landed
        } else {
            asm volatile("s_wait_asynccnt 0" ::: "memory");  // drain
        }

        // Read this wave's 512B row back from LDS (ds_load_b128 per lane).
        const float* src = &buf[b][wv][lane * 4];
        const float x0 = src[0], x1 = src[1], x2 = src[2], x3 = src[3];

        // Segment id is wave-uniform: force it onto the scalar path so the
        // fetch becomes an SMEM load and the atomics get a scalar base.
        const int urow = __builtin_amdgcn_readfirstlane(row);
        const int seg  = idx[urow];
        float* dst = out + (size_t)seg * DCOLS + lane * 4;

        // Fire-and-forget f32 atomics; destination region (25.6MB) lives in L2.
        atomicAdd(dst + 0, x0);
        atomicAdd(dst + 1, x1);
        atomicAdd(dst + 2, x2);
        atomicAdd(dst + 3, x3);

        row = nxt;
        b ^= 1;
    }
}

__global__ __launch_bounds__(256) void AggrSum_zero_kernel(floatx4* __restrict__ out, int n4) {
    int i = blockIdx.x * blockDim.x + threadIdx.x;
    if (i < n4) {
        floatx4 z = {0.0f, 0.0f, 0.0f, 0.0f};
        out[i] = z;
    }
}

extern "C" void kernel_launch(void* const* d_in, const int* in_sizes, int n_in,
                              void* d_out, int out_size, void* d_ws, size_t ws_size,
                              hipStream_t stream) {
    const float* H   = (const float*)d_in[0];
    const int*   idx = (const int*)d_in[1];
    float*       out = (float*)d_out;

    const int nrows = in_sizes[1];        // 625000
    const int n4    = out_size / 4;       // 50000*128/4 float4 stores

    // Output accumulates via atomics -> must be zeroed on every launch.
    AggrSum_zero_kernel<<<(n4 + 255) / 256, 256, 0, stream>>>((floatx4*)out, n4);

    // 8192 blocks x 8 waves = 65536 waves; ~9.5 rows per wave (grid-stride).
    AggrSum_scatter_kernel<<<8192, BLOCK, 0, stream>>>(H, idx, out, nrows);
}